// CASCADESAdapter_68977174774437
// MI455X (gfx1250) — compile-verified
//
#include <hip/hip_runtime.h>
#include <math.h>

typedef float v2f __attribute__((ext_vector_type(2)));
typedef float v8f __attribute__((ext_vector_type(8)));

#define B_ 4
#define S_ 2048
#define D_ 4096
#define R_ 8
#define K_ 4

// workspace layout (in floats)
#define CENT_OFF 0
#define W_OFF    (B_ * D_)                 // 16384
#define Z_OFF    (W_OFF + B_ * R_ * D_)    // 16384 + 131072

// ---------------------------------------------------------------------------
// K1: centroid[b,d] = 0.3/S * sum_s x[b,s,d] + 0.7 * x[b,S-1,d]
// grid = B * (D/256) * 8 s-chunks, block = 256
// ---------------------------------------------------------------------------
__global__ void __launch_bounds__(256) k_centroid(const float* __restrict__ x,
                                                  float* __restrict__ cent) {
  int blk  = blockIdx.x;
  int sBlk = blk & 7;          // 8 s-chunks of 256
  int dBlk = (blk >> 3) & 15;  // 16 d-chunks of 256
  int b    = blk >> 7;
  int d    = dBlk * 256 + threadIdx.x;
  const float* xp = x + (size_t)b * S_ * D_ + d;
  int s0 = sBlk * 256;
  float acc = 0.f;
#pragma unroll 4
  for (int s = 0; s < 256; ++s) acc += xp[(size_t)(s0 + s) * D_];
  float val = acc * (0.3f / (float)S_);
  if (sBlk == 7) val += 0.7f * xp[(size_t)(S_ - 1) * D_];
  atomicAdd(cent + b * D_ + d, val);
}

// ---------------------------------------------------------------------------
// K2: routing + gate, then W[b][i][d] = gate * sum_j Lam[b][i][j] * V[j][d]
// single block of 256 threads (all data tiny / L2-resident)
// ---------------------------------------------------------------------------
__global__ void __launch_bounds__(256) k_route(const float* __restrict__ cent,
                                               const float* __restrict__ U,
                                               const float* __restrict__ V,
                                               const float* __restrict__ core_pool,
                                               const float* __restrict__ core_keys,
                                               const float* __restrict__ gate_w,
                                               const float* __restrict__ gate_b,
                                               float* __restrict__ Wd) {
  __shared__ float red[256];
  __shared__ float lam[B_ * R_ * R_];  // gate-scaled Lam
  __shared__ float s_cn2[B_], s_kn2[K_], s_dot[B_][K_], s_gu[R_], s_gv[R_];
  int t = threadIdx.x;

  auto blockReduce = [&](float v) -> float {
    red[t] = v;
    __syncthreads();
    for (int off = 128; off > 0; off >>= 1) {
      if (t < off) red[t] += red[t + off];
      __syncthreads();
    }
    float r = red[0];
    __syncthreads();
    return r;
  };

  for (int b = 0; b < B_; ++b) {
    float a = 0.f;
    for (int d = t; d < D_; d += 256) { float c = cent[b * D_ + d]; a += c * c; }
    float r = blockReduce(a);
    if (t == 0) s_cn2[b] = r;
  }
  for (int k = 0; k < K_; ++k) {
    float a = 0.f;
    for (int d = t; d < D_; d += 256) { float c = core_keys[k * D_ + d]; a += c * c; }
    float r = blockReduce(a);
    if (t == 0) s_kn2[k] = r;
  }
  for (int b = 0; b < B_; ++b)
    for (int k = 0; k < K_; ++k) {
      float a = 0.f;
      for (int d = t; d < D_; d += 256) a += cent[b * D_ + d] * core_keys[k * D_ + d];
      float r = blockReduce(a);
      if (t == 0) s_dot[b][k] = r;
    }
  for (int r = 0; r < R_; ++r) {
    float a = 0.f;
    for (int d = t; d < D_; d += 256) a += U[d * R_ + r];
    float rr = blockReduce(a);
    if (t == 0) s_gu[r] = rr / (float)D_;
  }
  for (int r = 0; r < R_; ++r) {
    float a = 0.f;
    for (int d = t; d < D_; d += 256) a += V[r * D_ + d];
    float rr = blockReduce(a);
    if (t == 0) s_gv[r] = rr / (float)D_;
  }
  __syncthreads();

  if (t == 0) {
    float gz = gate_b[0];
    for (int r = 0; r < R_; ++r) gz += s_gu[r] * gate_w[r] + s_gv[r] * gate_w[R_ + r];
    float gate = 1.f / (1.f + expf(-gz));
    for (int b = 0; b < B_; ++b) {
      float w[K_]; float mx = -1e30f;
      float cn = fmaxf(sqrtf(s_cn2[b]), 1e-8f);
      for (int k = 0; k < K_; ++k) {
        float kn = fmaxf(sqrtf(s_kn2[k]), 1e-8f);
        w[k] = (s_dot[b][k] / (cn * kn)) * 20.0f;  // /TEMP(0.05)
        mx = fmaxf(mx, w[k]);
      }
      float sum = 0.f;
      for (int k = 0; k < K_; ++k) { w[k] = expf(w[k] - mx); sum += w[k]; }
      for (int k = 0; k < K_; ++k) w[k] /= sum;
      for (int i = 0; i < R_; ++i)
        for (int j = 0; j < R_; ++j) {
          float a = 0.f;
          for (int k = 0; k < K_; ++k) a += w[k] * core_pool[k * R_ * R_ + i * R_ + j];
          lam[(b * R_ + i) * R_ + j] = gate * a;
        }
    }
  }
  __syncthreads();

  for (int idx = t; idx < B_ * R_ * D_; idx += 256) {
    int d  = idx & (D_ - 1);
    int bi = idx >> 12;
    int i  = bi & (R_ - 1);
    int b  = bi >> 3;
    float a = 0.f;
#pragma unroll
    for (int j = 0; j < R_; ++j) a += lam[(b * R_ + i) * R_ + j] * V[j * D_ + d];
    Wd[idx] = a;
  }
}

// ---------------------------------------------------------------------------
// K3: Z[b,s,0..7] += x_b @ W_b^T over one K-chunk (K-split x4 for occupancy)
// grid = B * 16 sGroups * 4 kChunks = 256 blocks x 256 threads (8 waves)
// each wave: one 16x16 WMMA tile over 1024 K values; partials via f32 atomics
// W chunk staged into LDS with GLOBAL_LOAD_ASYNC_TO_LDS_B128 (ASYNCcnt path)
// ---------------------------------------------------------------------------
#define KQ_     4
#define KCHUNK  (D_ / KQ_)   // 1024
#define LSTRIDE 1028         // +4 floats: bank-conflict padding, keeps 16B align
#define ZROW    8            // broadcast zero row for N columns 8..15

__global__ void __launch_bounds__(256) k_xw(const float* __restrict__ x,
                                            const float* __restrict__ Wd,
                                            float* __restrict__ Z) {
  __shared__ float lw[(R_ + 1) * LSTRIDE];  // 9*1028*4 = 37008 bytes
  int blk = blockIdx.x;
  int kq  = blk & 3;
  int sG  = (blk >> 2) & 15;
  int b   = blk >> 6;
  int kc  = kq * KCHUNK;

  // zero the broadcast row (read by lanes whose N column is padding)
  for (int c2 = threadIdx.x; c2 < LSTRIDE; c2 += 256)
    lw[ZROW * LSTRIDE + c2] = 0.f;

  // async-stage W[b][0..7][kc..kc+KCHUNK) into padded LDS (16B per op)
  for (int q = threadIdx.x; q < R_ * (KCHUNK / 4); q += 256) {
    int row  = q >> 8;        // q / 256
    int col4 = q & 255;
    unsigned long long gaddr = (unsigned long long)(uintptr_t)
        (Wd + (size_t)(b * R_ + row) * D_ + kc + col4 * 4);
    unsigned int laddr = (unsigned int)(uintptr_t)(&lw[row * LSTRIDE + col4 * 4]);
    asm volatile("global_load_async_to_lds_b128 %0, %1, off"
                 :: "v"(laddr), "v"(gaddr) : "memory");
  }
  asm volatile("s_wait_asynccnt 0" ::: "memory");
  __syncthreads();

  int wave = threadIdx.x >> 5;
  int lane = threadIdx.x & 31;
  int m  = lane & 15;              // A-row within tile
  int hi = lane >> 4;              // K-half selector
  int n  = lane & 15;              // B/N column within tile
  int rowSel = (n < 8) ? n : ZROW; // padding columns read the zero row

  int sRow = sG * 128 + wave * 16 + m;
  const float* xrow = x + (size_t)(b * S_ + sRow) * D_ + kc + 2 * hi;
  const float* brow = &lw[rowSel * LSTRIDE + 2 * hi];

  v8f c = {};
#pragma unroll 4
  for (int j = 0; j < KCHUNK / 4; ++j) {
    v2f a  = *(const v2f*)(xrow + 4 * j);   // A: 16x4 of x (per-lane contiguous)
    v2f bb = *(const v2f*)(brow + 4 * j);   // B: 4x16 of W^T from LDS
    c = __builtin_amdgcn_wmma_f32_16x16x4_f32(false, a, false, bb,
                                              (short)0, c, false, false);
  }

  // accumulate partial tile into Z[B,S,16] (only the 8 real columns)
  if (n < 8) {
    float* zrow = Z + (size_t)(b * S_ + sG * 128 + wave * 16 + hi * 8) * 16 + n;
#pragma unroll
    for (int v = 0; v < 8; ++v) atomicAdd(zrow + (size_t)v * 16, c[v]);
  }
}

// ---------------------------------------------------------------------------
// K4: out[b,s,o] = sum_i Z[b,s,i] * U[o,i]   (gate already folded into W)
// one 16x16 output tile per wave, 2 WMMAs; pure store-bandwidth kernel
// grid = 16384 blocks x 256 threads (8 waves)
// ---------------------------------------------------------------------------
__global__ void __launch_bounds__(256) k_expand(const float* __restrict__ Z,
                                                const float* __restrict__ U,
                                                float* __restrict__ out) {
  int tile  = blockIdx.x * 8 + (threadIdx.x >> 5);
  int lane  = threadIdx.x & 31;
  int oTile = tile & 255;   // 256 o-tiles (D/16)
  int sTile = tile >> 8;    // 512 s-tiles (B*S/16)
  int b  = sTile >> 7;
  int sb = (sTile & 127) << 4;
  int m = lane & 15, hi = lane >> 4, n = lane & 15;

  const float* zrow = Z + (size_t)(b * S_ + sb + m) * 16 + 2 * hi;
  const float* urow = U + (size_t)(oTile * 16 + n) * R_ + 2 * hi;

  v8f c = {};
#pragma unroll
  for (int j = 0; j < 2; ++j) {
    v2f a  = *(const v2f*)(zrow + 4 * j);   // A: 16x4 of Z
    v2f bb = *(const v2f*)(urow + 4 * j);   // B: 4x16 of U^T
    c = __builtin_amdgcn_wmma_f32_16x16x4_f32(false, a, false, bb,
                                              (short)0, c, false, false);
  }
  float* orow = out + (size_t)(b * S_ + sb + hi * 8) * D_ + oTile * 16 + n;
#pragma unroll
  for (int v = 0; v < 8; ++v) orow[(size_t)v * D_] = c[v];
}

// ---------------------------------------------------------------------------
extern "C" void kernel_launch(void* const* d_in, const int* in_sizes, int n_in,
                              void* d_out, int out_size, void* d_ws, size_t ws_size,
                              hipStream_t stream) {
  (void)in_sizes; (void)n_in; (void)out_size; (void)ws_size;
  const float* x         = (const float*)d_in[0];
  const float* U         = (const float*)d_in[1];
  const float* V         = (const float*)d_in[2];
  const float* core_pool = (const float*)d_in[3];
  const float* core_keys = (const float*)d_in[4];
  const float* gate_w    = (const float*)d_in[5];
  const float* gate_b    = (const float*)d_in[6];

  float* ws   = (float*)d_ws;
  float* cent = ws + CENT_OFF;
  float* Wd   = ws + W_OFF;
  float* Z    = ws + Z_OFF;

  hipMemsetAsync(cent, 0, B_ * D_ * sizeof(float), stream);
  hipMemsetAsync(Z, 0, (size_t)B_ * S_ * 16 * sizeof(float), stream);
  k_centroid<<<B_ * 16 * 8, 256, 0, stream>>>(x, cent);
  k_route<<<1, 256, 0, stream>>>(cent, U, V, core_pool, core_keys, gate_w, gate_b, Wd);
  k_xw<<<B_ * 16 * KQ_, 256, 0, stream>>>(x, Wd, Z);
  k_expand<<<(B_ * S_ / 16) * (D_ / 16) / 8, 256, 0, stream>>>(Z, U, (float*)d_out);
}